// VerticalComparisonModelForBlockA_84593675862346
// MI455X (gfx1250) — compile-verified
//
#include <hip/hip_runtime.h>

typedef __attribute__((ext_vector_type(2))) float v2f;
typedef __attribute__((ext_vector_type(8))) float v8f;

#define NN 64
#define DD 128
#define LSTRIDE 130          // 128 + 2 pad: keeps float2 align, conflict-free pair pass
#define NSP 128              // S*P blocks
#define TPB 256              // 8 waves (wave32)
#define EPSN 1e-12f

__global__ __launch_bounds__(TPB)
void vcmp_fused_kernel(const float* __restrict__ xg,
                       const float* __restrict__ yg,
                       float* __restrict__ out) {
  extern __shared__ float smem[];
  float* invnx = smem;                    // [64]
  float* invny = smem + 64;               // [64]
  float* lx    = smem + 128;              // [64*130]
  float* ly    = lx + NN * LSTRIDE;       // [64*130]

  const int tid = threadIdx.x;
  const int blk = blockIdx.x;             // (s,p) block, 0..127
  const int role = blockIdx.y;            // 0: wmma+cos, 1: l1/l2
  const float* xb = xg + (size_t)blk * NN * DD;
  const float* yb = yg + (size_t)blk * NN * DD;
  float* ob = out + (size_t)blk * NN * NN * 3;

  // ---- Phase 1: stage x,y into padded LDS (coalesced float2) ----
  const v2f* xg2 = (const v2f*)xb;
  const v2f* yg2 = (const v2f*)yb;
  for (int idx = tid; idx < NN * DD / 2; idx += TPB) {
    int row = idx >> 6;                   // DD/2 = 64 float2 per row
    int c2  = idx & 63;
    *(v2f*)(lx + row * LSTRIDE + 2 * c2) = xg2[idx];
    *(v2f*)(ly + row * LSTRIDE + 2 * c2) = yg2[idx];
  }
  __syncthreads();

  if (role == 0) {
    // ---- Phase 2: row norms -> 1/max(||r||, eps) ----
    if (tid < 2 * NN) {
      const float* rowp = (tid < NN) ? (lx + tid * LSTRIDE)
                                     : (ly + (tid - NN) * LSTRIDE);
      float ss = 0.f;
      #pragma unroll 8
      for (int d = 0; d < DD; ++d) { float v = rowp[d]; ss += v * v; }
      float inv = 1.0f / fmaxf(sqrtf(ss), EPSN);
      if (tid < NN) invnx[tid] = inv; else invny[tid - NN] = inv;
    }
    __syncthreads();

    // ---- Phase 3: gram matrix via V_WMMA_F32_16X16X4_F32, scaled -> cos ----
    const int wave = tid >> 5;
    const int lane = tid & 31;
    const int h = lane >> 4;              // half-wave: K pair select / M+8 select
    const int r = lane & 15;
    for (int t = wave; t < 16; t += 8) {  // 16 tiles, 2 per wave
      const int tm = t >> 2, tn = t & 3;
      const float* ax = lx + (tm * 16 + r) * LSTRIDE + 2 * h;  // A[M=r][K=2h..2h+1]
      const float* by = ly + (tn * 16 + r) * LSTRIDE + 2 * h;  // B[K][N=r] from row-major y
      v8f acc = {0.f, 0.f, 0.f, 0.f, 0.f, 0.f, 0.f, 0.f};
      #pragma unroll
      for (int k0 = 0; k0 < DD; k0 += 4) {
        v2f a = *(const v2f*)(ax + k0);
        v2f b = *(const v2f*)(by + k0);
        acc = __builtin_amdgcn_wmma_f32_16x16x4_f32(
            /*neg_a=*/false, a, /*neg_b=*/false, b,
            /*c_mod=*/(short)0, acc, /*reuse_a=*/false, /*reuse_b=*/false);
      }
      const int m = tn * 16 + r;
      const float inm = invny[m];
      #pragma unroll
      for (int v = 0; v < 8; ++v) {       // C/D layout: VGPR v, lane half h -> M = v + 8h
        const int n = tm * 16 + v + 8 * h;
        ob[(n * NN + m) * 3 + 0] = acc[v] * invnx[n] * inm;
      }
    }
  } else {
    // ---- Phase 4: exact L2 (cdist p=2) and L1 (cdist p=1) ----
    for (int i = 0; i < 16; ++i) {
      const int pair = i * TPB + tid;     // lanes -> consecutive m: conflict-free banks
      const int n = pair >> 6;
      const int m = pair & 63;
      const v2f* xr = (const v2f*)(lx + n * LSTRIDE);
      const v2f* yr = (const v2f*)(ly + m * LSTRIDE);
      float l1 = 0.f, sq = 0.f;
      #pragma unroll 8
      for (int d2 = 0; d2 < DD / 2; ++d2) {
        v2f fx = xr[d2], fy = yr[d2];
        float dx = fx.x - fy.x, dy = fx.y - fy.y;
        l1 += fabsf(dx) + fabsf(dy);
        sq += dx * dx + dy * dy;
      }
      ob[pair * 3 + 1] = sqrtf(sq);
      ob[pair * 3 + 2] = l1;
    }
  }
}

extern "C" void kernel_launch(void* const* d_in, const int* in_sizes, int n_in,
                              void* d_out, int out_size, void* d_ws, size_t ws_size,
                              hipStream_t stream) {
  const float* x = (const float*)d_in[0];
  const float* y = (const float*)d_in[1];
  float* out = (float*)d_out;
  size_t shmem = (size_t)(128 + 2 * NN * LSTRIDE) * sizeof(float);  // ~67 KB dynamic LDS
  dim3 grid(NSP, 2);
  vcmp_fused_kernel<<<grid, TPB, shmem, stream>>>(x, y, out);
}